// ContactLoss_41721312313535
// MI455X (gfx1250) — compile-verified
//
#include <hip/hip_runtime.h>
#include <stdint.h>

#define THRESH_C  0.001f
#define RADIUS_C  0.1f
#define NSAMPLE_C 8
#define EPS_C     1e-07f

#define TPB 256            // threads per block (8 waves of 32)
#define PPB 128            // points per block (16 points per wave)
#define CHUNK_TILES 64     // 16x16 face tiles staged per chunk
#define CHUNKF (CHUNK_TILES * 16)          // faces per chunk (1024)
#define STAGE_PER_THREAD (CHUNKF / TPB)    // float4 copies per thread (4)

typedef float v2f __attribute__((ext_vector_type(2)));
typedef float v8f __attribute__((ext_vector_type(8)));

#ifndef __has_builtin
#define __has_builtin(x) 0
#endif

#if __has_builtin(__builtin_amdgcn_wmma_f32_16x16x4_f32)
#define HAVE_WMMA_F32X4 1
#else
#define HAVE_WMMA_F32X4 0
#endif

static __device__ __forceinline__ unsigned ballot32(bool p) {
#if __has_builtin(__builtin_amdgcn_ballot_w32)
  return __builtin_amdgcn_ballot_w32(p);
#else
  return (unsigned)__ballot(p);
#endif
}

// Wait for all outstanding async (global<->LDS) transfers of this wave.
static __device__ __forceinline__ void wait_async0() {
#if __has_builtin(__builtin_amdgcn_s_wait_asynccnt)
  __builtin_amdgcn_s_wait_asynccnt(0);
#else
  asm volatile("s_wait_asynccnt 0x0" ::: "memory");
#endif
}

// Async DMA: 16 bytes global -> LDS per lane (ASYNCcnt-tracked).
// VDST operand = 32-bit LDS address (low 32 bits of generic LDS pointer:
// generic-LDS addr = {aperture, lds_offset[31:0]}).
static __device__ __forceinline__ void async_g2l_b128(const float4* gsrc,
                                                      float4* ldst) {
  unsigned lds_addr = (unsigned)(uintptr_t)ldst;
  asm volatile("global_load_async_to_lds_b128 %0, %1, off"
               :
               : "v"(lds_addr), "v"(gsrc)
               : "memory");
}

// ---------------------------------------------------------------------------
// Kernel 1: per-face prep.
//   fp4[b][f]    = { -2*c0.x, -2*c0.y, -2*c0.z, |c0|^2 }  (t0 center, packed
//                   so WMMA B-matrix lanes load 8B at +hi*8)
//   fpos4[b][f]  = { c1.x, c1.y, c1.z, 0 }                 (t1 center)
//   fnorm4[b][f] = { n.x,  n.y,  n.z,  0 }                 (t1 normal)
// Padding faces (f in [F, Fpad)) get |c0|^2 = +huge so they never hit.
// ---------------------------------------------------------------------------
__global__ void face_prep_kernel(const float* __restrict__ h_state,
                                 const int*   __restrict__ h_faces,
                                 float* __restrict__ fp4,
                                 float* __restrict__ fpos4,
                                 float* __restrict__ fnorm4,
                                 int M, int F, int Fpad)
{
  const int b = blockIdx.y;
  const int f = blockIdx.x * blockDim.x + threadIdx.x;
  if (f >= Fpad) return;
  const size_t fo = ((size_t)b * Fpad + f) * 4;
  if (f >= F) {
    fp4[fo + 0] = 0.f; fp4[fo + 1] = 0.f; fp4[fo + 2] = 0.f;
    fp4[fo + 3] = 3.0e38f;
    fpos4[fo + 0] = 0.f; fpos4[fo + 1] = 0.f; fpos4[fo + 2] = 0.f; fpos4[fo + 3] = 0.f;
    fnorm4[fo + 0] = 0.f; fnorm4[fo + 1] = 0.f; fnorm4[fo + 2] = 0.f; fnorm4[fo + 3] = 0.f;
    return;
  }
  const int* fc = h_faces + ((size_t)b * F + f) * 3;
  const int v0 = fc[0], v1 = fc[1], v2 = fc[2];
  const float* hs = h_state + (size_t)b * M * 9;
  const float* p0 = hs + (size_t)v0 * 9;
  const float* p1 = hs + (size_t)v1 * 9;
  const float* p2 = hs + (size_t)v2 * 9;

  const float ax = p0[0], ay = p0[1], az = p0[2];
  const float bx = p1[0], by = p1[1], bz = p1[2];
  const float cx = p2[0], cy = p2[1], cz = p2[2];
  const float k3 = 1.0f / 3.0f;
  const float c1x = (ax + bx + cx) * k3;
  const float c1y = (ay + by + cy) * k3;
  const float c1z = (az + bz + cz) * k3;
  const float e1x = bx - ax, e1y = by - ay, e1z = bz - az;
  const float e2x = cx - ax, e2y = cy - ay, e2z = cz - az;
  float nx = e1y * e2z - e1z * e2y;
  float ny = e1z * e2x - e1x * e2z;
  float nz = e1x * e2y - e1y * e2x;
  const float nn  = sqrtf(nx * nx + ny * ny + nz * nz);
  const float inv = 1.0f / fmaxf(nn, EPS_C);
  nx *= inv; ny *= inv; nz *= inv;
  const float c0x = (p0[6] + p1[6] + p2[6]) * k3;
  const float c0y = (p0[7] + p1[7] + p2[7]) * k3;
  const float c0z = (p0[8] + p1[8] + p2[8]) * k3;

  fp4[fo + 0] = -2.0f * c0x;
  fp4[fo + 1] = -2.0f * c0y;
  fp4[fo + 2] = -2.0f * c0z;
  fp4[fo + 3] = c0x * c0x + c0y * c0y + c0z * c0z;

  fpos4[fo + 0] = c1x; fpos4[fo + 1] = c1y; fpos4[fo + 2] = c1z; fpos4[fo + 3] = 0.f;
  fnorm4[fo + 0] = nx; fnorm4[fo + 1] = ny; fnorm4[fo + 2] = nz; fnorm4[fo + 3] = 0.f;
}

// ---------------------------------------------------------------------------
// Kernel 2: ball-query scan (WMMA distance tiles, async-LDS staged faces,
//           software-pipelined B loads) + contact loss.
// ---------------------------------------------------------------------------
__global__ void contact_scan_kernel(const float* __restrict__ pred,
                                    const float* __restrict__ state,
                                    const float* __restrict__ fp4,
                                    const float* __restrict__ fpos4,
                                    const float* __restrict__ fnorm4,
                                    float* __restrict__ partials,
                                    int N, int Fpad)
{
  __shared__ float4 s_stage[2][CHUNKF];   // double-buffered face chunks (32 KB)
  __shared__ int    s_cnt[PPB];
  __shared__ int    s_list[PPB][NSAMPLE_C];
  __shared__ float  s_xsq[PPB];
#if !HAVE_WMMA_F32X4
  __shared__ float  s_px[PPB], s_py[PPB], s_pz[PPB];
#endif
  __shared__ float  s_red[TPB];

  const int b    = blockIdx.y;
  const int tid  = threadIdx.x;
  const int wave = tid >> 5;
  const int lane = tid & 31;
  const int hi   = lane >> 4;
  const int pl   = lane & 15;
  const int wave_base = blockIdx.x * PPB + wave * 16;
  const int slot_base = wave * 16;

  if (tid < PPB) { s_cnt[tid] = 0; s_list[tid][0] = 0; }

  // ---- load this lane's point (t0 coords), clamped for tail waves ----
  int pidx = wave_base + pl;
  if (pidx > N - 1) pidx = N - 1;
  const float* xp = state + ((size_t)b * N + pidx) * 3;
  const float x0 = xp[0], x1 = xp[1], x2 = xp[2];
  const float xsq = x0 * x0 + x1 * x1 + x2 * x2;
  s_xsq[slot_base + pl] = xsq;            // lanes l and l+16 write same value
#if !HAVE_WMMA_F32X4
  s_px[slot_base + pl] = x0; s_py[slot_base + pl] = x1; s_pz[slot_base + pl] = x2;
#endif

  const float4* fb4 = (const float4*)(fp4 + (size_t)b * Fpad * 4);
  const int nchunks = Fpad / CHUNKF;

  // stage chunk 0
#pragma unroll
  for (int k = 0; k < STAGE_PER_THREAD; ++k) {
    const int idx = tid + k * TPB;
    async_g2l_b128(fb4 + idx, &s_stage[0][idx]);
  }
  wait_async0();
  __syncthreads();

  // per-row hit thresholds: row m of the D tile is point (slot_base + m)
  const float r2 = RADIUS_C * RADIUS_C;
  float thr[8];
#pragma unroll
  for (int r = 0; r < 8; ++r)
    thr[r] = r2 - s_xsq[slot_base + hi * 8 + r];

  // per-lane byte offset of this lane's 8-byte B fragment within a tile row
  const int boff = pl * 16 + hi * 8;

#if HAVE_WMMA_F32X4
  // A matrix 16x4 = [x0 x1 x2 1]: lanes 0-15 hold K=0,1 ; lanes 16-31 K=2,3
  v2f amat;
  amat.x = hi ? x2   : x0;
  amat.y = hi ? 1.0f : x1;
#endif

  for (int c = 0; c < nchunks; ++c) {
    const int sel = c & 1;
    // prefetch next chunk into the other buffer while we consume this one
    if (c + 1 < nchunks) {
      const float4* src = fb4 + (size_t)(c + 1) * CHUNKF;
#pragma unroll
      for (int k = 0; k < STAGE_PER_THREAD; ++k) {
        const int idx = tid + k * TPB;
        async_g2l_b128(src + idx, &s_stage[sel ^ 1][idx]);
      }
    }

    const char* sb = (const char*)&s_stage[sel][0];

    // software pipeline: fetch B for tile t+1 while tile t is in the WMMA.
    // Prefetch never crosses the chunk boundary (other buffer is in flight).
    float2 bcur = *(const float2*)(sb + boff);
#pragma unroll 2
    for (int t = 0; t < CHUNK_TILES; ++t) {
      const int tn = (t + 1 < CHUNK_TILES) ? (t + 1) : t;   // clamped, branchless
      const float2 bnext = *(const float2*)(sb + (size_t)tn * 256 + boff);

      float d[8];
#if HAVE_WMMA_F32X4
      v2f bmat; bmat.x = bcur.x; bmat.y = bcur.y;
      v8f cz = {0.f, 0.f, 0.f, 0.f, 0.f, 0.f, 0.f, 0.f};
      v8f dd = __builtin_amdgcn_wmma_f32_16x16x4_f32(
          false, amat, false, bmat, (short)0, cz, false, false);
#pragma unroll
      for (int r = 0; r < 8; ++r) d[r] = dd[r];
#else
      const float* q = (const float*)(sb + (size_t)t * 256) + pl * 4;
      const float q0 = q[0], q1 = q[1], q2 = q[2], q3 = q[3];
#pragma unroll
      for (int r = 0; r < 8; ++r) {
        const int m = slot_base + hi * 8 + r;
        d[r] = q0 * s_px[m] + q1 * s_py[m] + q2 * s_pz[m] + q3;
      }
#endif
      // fast path: one ballot over min_r(d[r]-thr[r]) < 0. With denormals
      // enabled, sign(d-thr) is exact, so this is equivalent to any(d<thr).
      float dm = d[0] - thr[0];
#pragma unroll
      for (int r = 1; r < 8; ++r) dm = fminf(dm, d[r] - thr[r]);

      if (ballot32(dm < 0.f)) {
        // rare path: exact per-row masks, lane-0 appends in ascending order
#pragma unroll
        for (int r = 0; r < 8; ++r) {
          const unsigned mask = ballot32(d[r] < thr[r]);
          if (mask) {
            if (lane == 0) {
              unsigned mm = mask;
              while (mm) {
                const int bpos = __builtin_ctz(mm);
                mm &= mm - 1;
                const int slot = slot_base + r + ((bpos >= 16) ? 8 : 0);
                const int cc = s_cnt[slot];
                if (cc < NSAMPLE_C) {
                  s_list[slot][cc] = (c * CHUNK_TILES + t) * 16 + (bpos & 15);
                  s_cnt[slot] = cc + 1;
                }
              }
            }
          }
        }
      }
      bcur = bnext;
    }

    wait_async0();     // next chunk fully in LDS (own wave's async ops)
    __syncthreads();   // ... and visible to / awaited by all waves
  }

  // pad empty slots with first hit (face 0 when no hits) — matches
  // reference where(valid, order, order[:, :1]) semantics
  if (tid < PPB) {
    const int cc = s_cnt[tid];
    const int f0 = s_list[tid][0];
    for (int s = cc; s < NSAMPLE_C; ++s) s_list[tid][s] = f0;
  }
  __syncthreads();

  // ---- loss: lane handles point pl of its wave; lanes split samples 4/4 ----
  const int  p2i   = wave_base + pl;
  const bool valid = p2i < N;
  const int  pc    = valid ? p2i : (N - 1);
  const float* yp  = pred + ((size_t)b * N + pc) * 3;
  const float y0 = yp[0], y1 = yp[1], y2 = yp[2];
  const float4* fpos = (const float4*)(fpos4 + (size_t)b * Fpad * 4);
  const float4* fnor = (const float4*)(fnorm4 + (size_t)b * Fpad * 4);
  const int slot = slot_base + pl;
  float acc = 0.f;
#pragma unroll
  for (int s = 0; s < 4; ++s) {
    const int f = s_list[slot][hi * 4 + s];
    const float4 P  = fpos[f];
    const float4 Nv = fnor[f];
    const float diff = (y0 - P.x) * Nv.x + (y1 - P.y) * Nv.y + (y2 - P.z) * Nv.z;
    float ti = THRESH_C - diff;
    ti = fmaxf(ti, 0.f);
    acc += ti * ti * ti;
  }
  if (!valid) acc = 0.f;

  // deterministic block tree-reduction
  s_red[tid] = acc;
  __syncthreads();
  for (int off = TPB / 2; off > 0; off >>= 1) {
    if (tid < off) s_red[tid] += s_red[tid + off];
    __syncthreads();
  }
  if (tid == 0)
    partials[(size_t)blockIdx.y * gridDim.x + blockIdx.x] = s_red[0];
}

// ---------------------------------------------------------------------------
// Kernel 3: deterministic sequential final reduction.
// ---------------------------------------------------------------------------
__global__ void finalize_kernel(const float* __restrict__ partials,
                                float* __restrict__ out, int n, float scale)
{
  if (blockIdx.x == 0 && threadIdx.x == 0) {
    float s = 0.f;
    for (int i = 0; i < n; ++i) s += partials[i];
    out[0] = s * scale;
  }
}

// ---------------------------------------------------------------------------
extern "C" void kernel_launch(void* const* d_in, const int* in_sizes, int n_in,
                              void* d_out, int out_size, void* d_ws, size_t ws_size,
                              hipStream_t stream) {
  const float* pred   = (const float*)d_in[0];
  const float* state  = (const float*)d_in[1];
  const float* hstate = (const float*)d_in[2];
  const int*   hfaces = (const int*)d_in[3];

  const int B = 2;
  const int N = in_sizes[0] / (B * 3);
  const int M = in_sizes[2] / (B * 9);
  const int F = in_sizes[3] / (B * 3);
  const int Fpad = (F + CHUNKF - 1) & ~(CHUNKF - 1);   // chunk-granular padding

  float* ws     = (float*)d_ws;
  float* fp4    = ws;
  float* fpos4  = fp4    + (size_t)B * Fpad * 4;
  float* fnorm4 = fpos4  + (size_t)B * Fpad * 4;
  float* parts  = fnorm4 + (size_t)B * Fpad * 4;

  dim3 fgrid((Fpad + 255) / 256, B);
  face_prep_kernel<<<fgrid, 256, 0, stream>>>(hstate, hfaces, fp4, fpos4, fnorm4,
                                              M, F, Fpad);

  const int gx = (N + PPB - 1) / PPB;
  dim3 sgrid(gx, B);
  contact_scan_kernel<<<sgrid, TPB, 0, stream>>>(pred, state, fp4, fpos4, fnorm4,
                                                 parts, N, Fpad);

  const float scale = 1.0f / ((float)N * (float)B);    // mean over N, mean over B
  finalize_kernel<<<1, 32, 0, stream>>>(parts, (float*)d_out, gx * B, scale);
}